// PerLayer_36172214567623
// MI455X (gfx1250) — compile-verified
//
#include <hip/hip_runtime.h>
#include <math.h>
#include <stdint.h>

// Problem constants (match reference)
#define N_NODES 50000
#define N_EDGES 800000
#define D_FEAT  64
#define K_STEPS 10
#define LAMBDA_F 3.0f

// ---------------------------------------------------------------------------
// Phase 0: degree counting (once per launch)
// ---------------------------------------------------------------------------
__global__ void k_zero_u32(unsigned* __restrict__ p, int n) {
    int i = blockIdx.x * blockDim.x + threadIdx.x;
    if (i < n) p[i] = 0u;
}

__global__ void k_count(const int* __restrict__ src, const int* __restrict__ dst,
                        unsigned* __restrict__ cnt) {
    int i = blockIdx.x * blockDim.x + threadIdx.x;
    if (i < 2 * N_EDGES) {
        int node = (i < N_EDGES) ? src[i] : dst[i - N_EDGES];
        atomicAdd(&cnt[node], 1u);
    }
}

// Single-block exclusive scan of counts -> offs, cursor copy, and
// dis[n] = (deg+1)^-1/2.  50000 elements, 49 tiles of 1024 — negligible cost.
__global__ void k_scan_deg(const unsigned* __restrict__ cnt,
                           unsigned* __restrict__ offs,
                           unsigned* __restrict__ cursor,
                           float* __restrict__ dis) {
    __shared__ unsigned sh[1024];
    int t = threadIdx.x;
    unsigned carry = 0u;
    for (int base = 0; base < N_NODES; base += 1024) {
        int i = base + t;
        unsigned v = (i < N_NODES) ? cnt[i] : 0u;
        sh[t] = v;
        __syncthreads();
        unsigned xv = v;
        for (int off = 1; off < 1024; off <<= 1) {
            unsigned yv = (t >= off) ? sh[t - off] : 0u;
            __syncthreads();
            xv += yv;
            sh[t] = xv;
            __syncthreads();
        }
        if (i < N_NODES) {
            unsigned excl = carry + xv - v;      // exclusive prefix
            offs[i]   = excl;
            cursor[i] = excl;
            dis[i]    = rsqrtf((float)v + 1.0f); // deg_inv_sqrt with self loop
        }
        carry += sh[1023];
        __syncthreads();
    }
    if (t == 0) offs[N_NODES] = carry; // == 2*N_EDGES
}

// Fill shared CSR: adjacency column index + signed incidence edge id.
__global__ void k_fill(const int* __restrict__ src, const int* __restrict__ dst,
                       unsigned* __restrict__ cursor,
                       unsigned* __restrict__ adj_col,
                       unsigned* __restrict__ inc_ent) {
    int i = blockIdx.x * blockDim.x + threadIdx.x;
    if (i >= 2 * N_EDGES) return;
    int e = (i < N_EDGES) ? i : i - N_EDGES;
    int s = src[e], d = dst[e];
    int row, col; bool plus;
    if (i < N_EDGES) { row = s; col = d; plus = (s >= d); }
    else             { row = d; col = s; plus = (d >  s); }
    unsigned pos = atomicAdd(&cursor[row], 1u);
    adj_col[pos] = (unsigned)col;
    inc_ent[pos] = (unsigned)e | (plus ? 0u : 0x80000000u);
}

// ---------------------------------------------------------------------------
// Hot loop kernels. One 16-lane half-wave per 64-float row (float4/lane).
// z-row gathers use the CDNA5 async Global->LDS engine with a 2-deep
// double buffer (8 KB LDS/block), retired with s_wait_asynccnt.
// ---------------------------------------------------------------------------
__device__ __forceinline__ void async_row_load(unsigned lds_addr, const void* gaddr) {
    asm volatile("global_load_async_to_lds_b128 %0, %1, off"
                 :: "v"(lds_addr), "v"((unsigned long long)gaddr) : "memory");
}

__global__ void k_node_yxbar(const float* __restrict__ x0,   // hh (original x)
                             const float* __restrict__ xc,   // current x_k
                             const float* __restrict__ zk,
                             const unsigned* __restrict__ offs,
                             const unsigned* __restrict__ adj_col,
                             const unsigned* __restrict__ inc_ent,
                             const float* __restrict__ dis,
                             float* __restrict__ y,
                             float* __restrict__ xbar,
                             int first) {
    __shared__ float4 zbuf[2][256];
    int tid = threadIdx.x;
    int gh = (blockIdx.x * blockDim.x + tid) >> 4;
    int l  = tid & 15;
    if (gh >= N_NODES) return;
    int n = gh;
    float dn = dis[n];
    unsigned b = offs[n], e_end = offs[n + 1];
    unsigned deg = e_end - b;

    // adj_mv = dn * ( dn*x[n] + sum dis[col]*x[col] )   (x is L2-resident)
    const float4* x4 = (const float4*)xc;
    float4 self = x4[(size_t)n * 16 + l];
    float4 acc  = make_float4(dn * self.x, dn * self.y, dn * self.z, dn * self.w);
    unsigned col_next = (deg > 0u) ? adj_col[b] : 0u;
    for (unsigned i = 0; i < deg; ++i) {
        unsigned col = col_next;
        if (i + 1 < deg) {
            col_next = adj_col[b + i + 1];
            __builtin_prefetch(&x4[(size_t)col_next * 16 + l], 0, 1); // global_prefetch_b8
        }
        float wc = dis[col];
        float4 v = x4[(size_t)col * 16 + l];
        acc.x += wc * v.x; acc.y += wc * v.y; acc.z += wc * v.z; acc.w += wc * v.w;
    }

    // incT_mv(z) = dn * sum(+/- z[e]) : async Global->LDS double-buffer pipeline
    float4 zacc = make_float4(0.f, 0.f, 0.f, 0.f);
    if (!first) {
        const float4* z4 = (const float4*)zk;
        unsigned lds0 = (unsigned)(uintptr_t)&zbuf[0][tid]; // slot stride = 4096 B
        unsigned ent_next = 0u;
        if (deg > 0u) {
            ent_next = inc_ent[b];
            async_row_load(lds0, &z4[(size_t)(ent_next & 0x7fffffffu) * 16 + l]);
        }
        for (unsigned k2 = 0; k2 < deg; ++k2) {
            unsigned ent = ent_next;
            if (k2 + 1 < deg) {
                ent_next = inc_ent[b + k2 + 1];
                unsigned slot = lds0 + (((k2 + 1) & 1u) << 12);
                async_row_load(slot, &z4[(size_t)(ent_next & 0x7fffffffu) * 16 + l]);
                asm volatile("s_wait_asynccnt 0x1" ::: "memory");
            } else {
                asm volatile("s_wait_asynccnt 0x0" ::: "memory");
            }
            float sgn = (ent & 0x80000000u) ? -1.0f : 1.0f;
            float4 zv = zbuf[k2 & 1u][tid];
            zacc.x += sgn * zv.x; zacc.y += sgn * zv.y;
            zacc.z += sgn * zv.z; zacc.w += sgn * zv.w;
        }
    }

    float4 h = ((const float4*)x0)[(size_t)n * 16 + l];
    float4 yv;
    yv.x = 0.25f * h.x + 0.75f * dn * acc.x;
    yv.y = 0.25f * h.y + 0.75f * dn * acc.y;
    yv.z = 0.25f * h.z + 0.75f * dn * acc.z;
    yv.w = 0.25f * h.w + 0.75f * dn * acc.w;
    ((float4*)y)[(size_t)n * 16 + l] = yv;

    float qdn = 0.25f * dn;
    float4 xb;
    xb.x = yv.x - qdn * zacc.x;
    xb.y = yv.y - qdn * zacc.y;
    xb.z = yv.z - qdn * zacc.z;
    xb.w = yv.w - qdn * zacc.w;
    ((float4*)xbar)[(size_t)n * 16 + l] = xb;
}

__global__ void k_edge_update(const int* __restrict__ src, const int* __restrict__ dst,
                              const float* __restrict__ dis,
                              const float* __restrict__ xbar,
                              float* __restrict__ zk, int first) {
    int gh = (blockIdx.x * blockDim.x + threadIdx.x) >> 4;
    int l  = threadIdx.x & 15;
    if (gh >= N_EDGES) return;
    int e = gh;
    int s = src[e], d = dst[e];
    int r = (s > d) ? s : d;
    int c = (s > d) ? d : s;
    float a = dis[r], bb = dis[c];

    const float4* xb4 = (const float4*)xbar;
    float4 vr = xb4[(size_t)r * 16 + l];
    float4 vc = xb4[(size_t)c * 16 + l];

    float4 z;
    if (first) z = make_float4(0.f, 0.f, 0.f, 0.f);
    else       z = ((const float4*)zk)[(size_t)e * 16 + l];

    z.x += 2.0f * (a * vr.x - bb * vc.x);
    z.y += 2.0f * (a * vr.y - bb * vc.y);
    z.z += 2.0f * (a * vr.z - bb * vc.z);
    z.w += 2.0f * (a * vr.w - bb * vc.w);

    // row L2 norm across the 16-lane half-wave
    float nn = z.x * z.x + z.y * z.y + z.z * z.z + z.w * z.w;
    nn += __shfl_xor(nn, 1, 16);
    nn += __shfl_xor(nn, 2, 16);
    nn += __shfl_xor(nn, 4, 16);
    nn += __shfl_xor(nn, 8, 16);
    float rn = sqrtf(nn);
    float scale = (rn > LAMBDA_F) ? (LAMBDA_F / rn) : 1.0f; // min(1, lambda/rn)
    z.x *= scale; z.y *= scale; z.z *= scale; z.w *= scale;

    ((float4*)zk)[(size_t)e * 16 + l] = z;
}

__global__ void k_node_xk(const float* __restrict__ y,
                          const float* __restrict__ zk,
                          const unsigned* __restrict__ offs,
                          const unsigned* __restrict__ inc_ent,
                          const float* __restrict__ dis,
                          float* __restrict__ xout) {
    __shared__ float4 zbuf[2][256];
    int tid = threadIdx.x;
    int gh = (blockIdx.x * blockDim.x + tid) >> 4;
    int l  = tid & 15;
    if (gh >= N_NODES) return;
    int n = gh;
    unsigned b = offs[n], e_end = offs[n + 1];
    unsigned deg = e_end - b;

    const float4* z4 = (const float4*)zk;
    float4 zacc = make_float4(0.f, 0.f, 0.f, 0.f);
    unsigned lds0 = (unsigned)(uintptr_t)&zbuf[0][tid];
    unsigned ent_next = 0u;
    if (deg > 0u) {
        ent_next = inc_ent[b];
        async_row_load(lds0, &z4[(size_t)(ent_next & 0x7fffffffu) * 16 + l]);
    }
    for (unsigned k2 = 0; k2 < deg; ++k2) {
        unsigned ent = ent_next;
        if (k2 + 1 < deg) {
            ent_next = inc_ent[b + k2 + 1];
            unsigned slot = lds0 + (((k2 + 1) & 1u) << 12);
            async_row_load(slot, &z4[(size_t)(ent_next & 0x7fffffffu) * 16 + l]);
            asm volatile("s_wait_asynccnt 0x1" ::: "memory");
        } else {
            asm volatile("s_wait_asynccnt 0x0" ::: "memory");
        }
        float sgn = (ent & 0x80000000u) ? -1.0f : 1.0f;
        float4 zv = zbuf[k2 & 1u][tid];
        zacc.x += sgn * zv.x; zacc.y += sgn * zv.y;
        zacc.z += sgn * zv.z; zacc.w += sgn * zv.w;
    }

    float dn = dis[n];
    float qdn = 0.25f * dn;
    float4 yv = ((const float4*)y)[(size_t)n * 16 + l];
    float4 xo;
    xo.x = yv.x - qdn * zacc.x;
    xo.y = yv.y - qdn * zacc.y;
    xo.z = yv.z - qdn * zacc.z;
    xo.w = yv.w - qdn * zacc.w;
    ((float4*)xout)[(size_t)n * 16 + l] = xo;
}

// ---------------------------------------------------------------------------
extern "C" void kernel_launch(void* const* d_in, const int* in_sizes, int n_in,
                              void* d_out, int out_size, void* d_ws, size_t ws_size,
                              hipStream_t stream) {
    const float* x   = (const float*)d_in[0];
    const int*   src = (const int*)d_in[1];
    const int*   dst = (const int*)d_in[2];

    char* ws = (char*)d_ws;
    size_t off = 0;
    auto take = [&](size_t bytes) -> char* {
        char* p = ws + off;
        off = (off + bytes + 255) & ~(size_t)255;
        return p;
    };
    unsigned* cnt     = (unsigned*)take((size_t)N_NODES * 4);
    unsigned* offs    = (unsigned*)take((size_t)(N_NODES + 1) * 4);
    unsigned* cursor  = (unsigned*)take((size_t)N_NODES * 4);
    float*    dis     = (float*)   take((size_t)N_NODES * 4);
    unsigned* adj_col = (unsigned*)take((size_t)2 * N_EDGES * 4);
    unsigned* inc_ent = (unsigned*)take((size_t)2 * N_EDGES * 4);
    float*    y       = (float*)   take((size_t)N_NODES * D_FEAT * 4);
    float*    xbar    = (float*)   take((size_t)N_NODES * D_FEAT * 4);
    float*    zk      = (float*)   take((size_t)N_EDGES * D_FEAT * 4);
    float*    xout    = (float*)d_out;

    // ---- build CSR (once per launch; no cached state) ----
    k_zero_u32<<<(N_NODES + 255) / 256, 256, 0, stream>>>(cnt, N_NODES);
    k_count<<<(2 * N_EDGES + 255) / 256, 256, 0, stream>>>(src, dst, cnt);
    k_scan_deg<<<1, 1024, 0, stream>>>(cnt, offs, cursor, dis);
    k_fill<<<(2 * N_EDGES + 255) / 256, 256, 0, stream>>>(src, dst, cursor, adj_col, inc_ent);

    // ---- 10 proximal steps, atomic-free gathers ----
    const int HWN = (N_NODES * 16 + 255) / 256; // half-wave per node row
    const int HWE = (N_EDGES * 16 + 255) / 256; // half-wave per edge row
    const float* xc = x;
    for (int k = 0; k < K_STEPS; ++k) {
        k_node_yxbar<<<HWN, 256, 0, stream>>>(x, xc, zk, offs, adj_col, inc_ent,
                                              dis, y, xbar, (k == 0) ? 1 : 0);
        k_edge_update<<<HWE, 256, 0, stream>>>(src, dst, dis, xbar, zk, (k == 0) ? 1 : 0);
        k_node_xk<<<HWN, 256, 0, stream>>>(y, zk, offs, inc_ent, dis, xout);
        xc = xout; // d_out doubles as the x_k ping buffer
    }
}